// InstanceLoss_28991029248339
// MI455X (gfx1250) — compile-verified
//
#include <hip/hip_runtime.h>
#include <hip/hip_bf16.h>
#include <math.h>

// ---------------------------------------------------------------------------
// Discriminative instance loss, MI455X (gfx1250, wave32).
// Segment-sum pass = one-hot GEMM on V_WMMA_F32_16X16X4_F32, widened to
// 32 pixels / wave-iteration (4x b128 emb + 2x... 8x b128 labels in flight)
// so each wave keeps ~2KB of HBM traffic outstanding -> saturates 23.3 TB/s.
// Counts ride on a per-lane VALU compare (lane id == cluster id), which
// co-executes with the XDL/WMMA pipe.
// ---------------------------------------------------------------------------

typedef __attribute__((ext_vector_type(2)))  float    v2f;
typedef __attribute__((ext_vector_type(8)))  float    v8f;

#define DELTA_VAR_  0.5f
#define DELTA_DIST_ 1.5f
#define GAMMA_      0.001f
#define NBATCH 4
#define EDIM   16
#define CPAD   32          // padded cluster count (real C = 20, keep = 1..19)

// workspace layout (floats)
#define OFF_SUMS   0       // [NBATCH][CPAD][EDIM] = 2048
#define OFF_COUNTS 2048    // [NBATCH][CPAD]       = 128
#define OFF_MEANS  2176    // [NBATCH][CPAD][EDIM] = 2048
#define OFF_HINGE  4224    // [NBATCH][CPAD]       = 128
#define OFF_PART   4352    // [NBATCH]             = 4
#define WS_FLOATS  4356

__global__ void __launch_bounds__(256)
zero_ws_kernel(float* __restrict__ ws, int n)
{
    int i = blockIdx.x * blockDim.x + threadIdx.x;
    if (i < n) ws[i] = 0.0f;
}

// ---------------------------------------------------------------------------
// Pass 1: segment sums via one-hot WMMA (counts via VALU).
// Tile = 32 pixels per wave-iteration. K-to-pixel map: group g (0..7),
// K index k (0..3) -> pixel base + g + 8k. Lane half hi holds K = 2hi,2hi+1,
// so each lane streams 16 contiguous floats at row + base + 16*hi (64B).
// P must be a multiple of 32 (400*800 = 320000 ok).
// ---------------------------------------------------------------------------
__global__ void __launch_bounds__(256)
seg_sum_kernel(const float* __restrict__ emb, const int* __restrict__ target,
               float* __restrict__ sums, float* __restrict__ counts, int P)
{
    const int n = blockIdx.y;
    const float* __restrict__ embn = emb + (size_t)n * EDIM * P;
    const int*   __restrict__ tn   = target + (size_t)n * P;

    const int lane  = threadIdx.x & 31;
    const int wave  = threadIdx.x >> 5;
    const int gwave = blockIdx.x * (blockDim.x >> 5) + wave;
    const int nwave = gridDim.x * (blockDim.x >> 5);

    const int m  = lane & 15;   // A row (cluster within chunk) == B/C column e
    const int e  = lane & 15;
    const int hi = lane >> 4;   // lane half -> K pair {2hi, 2hi+1}

    // even/odd-group accumulator pairs to shorten the WMMA dep chain
    v8f acc0a = {}, acc0b = {}, acc1a = {}, acc1b = {};
    int cntv = 0;               // count for cluster == lane (0..31)

    const float* __restrict__ rowp = embn + (size_t)e * P + 16 * hi;

    for (int base = gwave * 32; base < P; base += nwave * 32) {
        // 32 labels, broadcast loads (same address across lanes)
        int labs[32];
        #pragma unroll
        for (int q = 0; q < 8; ++q) {
            const int4 t4 = *reinterpret_cast<const int4*>(tn + base + q * 4);
            labs[q * 4 + 0] = t4.x; labs[q * 4 + 1] = t4.y;
            labs[q * 4 + 2] = t4.z; labs[q * 4 + 3] = t4.w;
        }
        // 64B contiguous per lane: pixels [base+16hi, base+16hi+16)
        float arr[16];
        #pragma unroll
        for (int q = 0; q < 4; ++q) {
            const float4 f4 = *reinterpret_cast<const float4*>(rowp + base + q * 4);
            arr[q * 4 + 0] = f4.x; arr[q * 4 + 1] = f4.y;
            arr[q * 4 + 2] = f4.z; arr[q * 4 + 3] = f4.w;
        }
        // counts: one compare per pixel against this lane's cluster id
        #pragma unroll
        for (int q = 0; q < 32; ++q) cntv += (labs[q] == lane) ? 1 : 0;

        // 8 groups x 2 cluster-chunks of accumulating WMMAs
        #pragma unroll
        for (int g = 0; g < 8; ++g) {
            const int la = labs[16 * hi + g];       // K = 2hi
            const int lb = labs[16 * hi + 8 + g];   // K = 2hi+1
            v2f a0, a1, b;
            a0.x = (la == m)      ? 1.0f : 0.0f;
            a0.y = (lb == m)      ? 1.0f : 0.0f;
            a1.x = (la == m + 16) ? 1.0f : 0.0f;
            a1.y = (lb == m + 16) ? 1.0f : 0.0f;
            b.x = arr[g];
            b.y = arr[8 + g];
            if (g & 1) {
                acc0b = __builtin_amdgcn_wmma_f32_16x16x4_f32(false, a0, false, b, (short)0, acc0b, false, false);
                acc1b = __builtin_amdgcn_wmma_f32_16x16x4_f32(false, a1, false, b, (short)0, acc1b, false, false);
            } else {
                acc0a = __builtin_amdgcn_wmma_f32_16x16x4_f32(false, a0, false, b, (short)0, acc0a, false, false);
                acc1a = __builtin_amdgcn_wmma_f32_16x16x4_f32(false, a1, false, b, (short)0, acc1a, false, false);
            }
        }
    }

    // Epilogue: C layout -> lane l, vgpr v holds (M = (l>=16?8:0)+v, N = l&15)
    atomicAdd(&counts[n * CPAD + lane], (float)cntv);
    #pragma unroll
    for (int v = 0; v < 8; ++v) {
        const int mrow = (hi << 3) + v;
        atomicAdd(&sums[((size_t)n * CPAD + mrow)      * EDIM + e], acc0a[v] + acc0b[v]);
        atomicAdd(&sums[((size_t)n * CPAD + 16 + mrow) * EDIM + e], acc1a[v] + acc1b[v]);
    }
}

// ---------------------------------------------------------------------------
// Pass 2 (tiny): means = sums/counts; push (distance) term + regularizer.
// ---------------------------------------------------------------------------
__global__ void __launch_bounds__(384)
means_dist_kernel(const float* __restrict__ sums, const float* __restrict__ counts,
                  float* __restrict__ means, float* __restrict__ partial)
{
    const int n = blockIdx.x;
    __shared__ float sm[CPAD * EDIM];
    __shared__ float accd, accr;
    if (threadIdx.x == 0) { accd = 0.0f; accr = 0.0f; }
    for (int t = threadIdx.x; t < CPAD * EDIM; t += blockDim.x) {
        const int   c   = t >> 4;
        const float cnt = counts[n * CPAD + c];
        const float mu  = (cnt > 0.0f) ? sums[(size_t)n * CPAD * EDIM + t] / cnt : 0.0f;
        sm[t] = mu;
        means[(size_t)n * CPAD * EDIM + t] = mu;
    }
    __syncthreads();
    const int t = threadIdx.x;
    if (t < 19) {                       // regularizer over keep clusters 1..19
        const int c = t + 1;
        float s = 0.0f;
        #pragma unroll
        for (int ee = 0; ee < EDIM; ++ee) { const float v = sm[c * EDIM + ee]; s = fmaf(v, v, s); }
        atomicAdd(&accr, (s > 0.0f) ? sqrtf(s) : 0.0f);
    }
    if (t < 361) {                      // 19x19 pairwise push term
        const int i = t / 19 + 1, j = t % 19 + 1;
        if (i != j) {
            float s = 0.0f;
            #pragma unroll
            for (int ee = 0; ee < EDIM; ++ee) {
                const float dv = sm[i * EDIM + ee] - sm[j * EDIM + ee];
                s = fmaf(dv, dv, s);
            }
            const float dm = (s > 0.0f) ? sqrtf(s) : 0.0f;
            const float h  = fmaxf(2.0f * DELTA_DIST_ - dm, 0.0f);
            atomicAdd(&accd, h * h);
        }
    }
    __syncthreads();
    if (t == 0)
        partial[n] = accd / (19.0f * 18.0f) + GAMMA_ * (accr / 19.0f);
}

// ---------------------------------------------------------------------------
// Pass 3: per-pixel pull (variance) hinge, 4 pixels/thread with float4 loads
// (17x b128 in flight per lane), reduced per cluster via LDS f32 atomics.
// P must be a multiple of 4.
// ---------------------------------------------------------------------------
__global__ void __launch_bounds__(256)
var_kernel(const float* __restrict__ emb, const int* __restrict__ target,
           const float* __restrict__ means, float* __restrict__ hinge, int P)
{
    __shared__ float smean[CPAD * EDIM];
    __shared__ float shinge[CPAD];
    const int n = blockIdx.y;
    for (int t = threadIdx.x; t < CPAD * EDIM; t += blockDim.x)
        smean[t] = means[(size_t)n * CPAD * EDIM + t];
    for (int t = threadIdx.x; t < CPAD; t += blockDim.x) shinge[t] = 0.0f;
    __syncthreads();

    const float* __restrict__ embn = emb + (size_t)n * EDIM * P;
    const int*   __restrict__ tn   = target + (size_t)n * P;
    const int tid    = blockIdx.x * blockDim.x + threadIdx.x;
    const int stride = gridDim.x * blockDim.x;

    for (int p4 = tid * 4; p4 < P; p4 += stride * 4) {
        const int4 lab = *reinterpret_cast<const int4*>(tn + p4);
        float d2x = 0.0f, d2y = 0.0f, d2z = 0.0f, d2w = 0.0f;
        #pragma unroll
        for (int ee = 0; ee < EDIM; ++ee) {
            const float4 v = *reinterpret_cast<const float4*>(embn + (size_t)ee * P + p4);
            const float* mc = smean + ee;
            float dv;
            dv = v.x - mc[lab.x * EDIM]; d2x = fmaf(dv, dv, d2x);
            dv = v.y - mc[lab.y * EDIM]; d2y = fmaf(dv, dv, d2y);
            dv = v.z - mc[lab.z * EDIM]; d2z = fmaf(dv, dv, d2z);
            dv = v.w - mc[lab.w * EDIM]; d2w = fmaf(dv, dv, d2w);
        }
        float h;
        if (lab.x != 0) { h = fmaxf(((d2x > 0.f) ? sqrtf(d2x) : 0.f) - DELTA_VAR_, 0.f); atomicAdd(&shinge[lab.x], h * h); }
        if (lab.y != 0) { h = fmaxf(((d2y > 0.f) ? sqrtf(d2y) : 0.f) - DELTA_VAR_, 0.f); atomicAdd(&shinge[lab.y], h * h); }
        if (lab.z != 0) { h = fmaxf(((d2z > 0.f) ? sqrtf(d2z) : 0.f) - DELTA_VAR_, 0.f); atomicAdd(&shinge[lab.z], h * h); }
        if (lab.w != 0) { h = fmaxf(((d2w > 0.f) ? sqrtf(d2w) : 0.f) - DELTA_VAR_, 0.f); atomicAdd(&shinge[lab.w], h * h); }
    }
    __syncthreads();
    for (int t = threadIdx.x; t < CPAD; t += blockDim.x)
        atomicAdd(&hinge[n * CPAD + t], shinge[t]);
}

// ---------------------------------------------------------------------------
// Pass 4: scalar assembly.
// ---------------------------------------------------------------------------
__global__ void __launch_bounds__(128)
finalize_kernel(const float* __restrict__ hinge, const float* __restrict__ counts,
                const float* __restrict__ partial, float* __restrict__ out)
{
    __shared__ float acc;
    if (threadIdx.x == 0) acc = 0.0f;
    __syncthreads();
    const int t = threadIdx.x;
    if (t < NBATCH * 19) {                  // pull term: hinge[c]/counts[c] / Ck
        const int n = t / 19, c = t % 19 + 1;
        atomicAdd(&acc, (hinge[n * CPAD + c] / counts[n * CPAD + c]) / 19.0f);
    }
    if (t < NBATCH) atomicAdd(&acc, partial[t]);
    __syncthreads();
    if (t == 0) out[0] = acc / (float)NBATCH;
}

extern "C" void kernel_launch(void* const* d_in, const int* in_sizes, int n_in,
                              void* d_out, int out_size, void* d_ws, size_t ws_size,
                              hipStream_t stream)
{
    (void)n_in; (void)out_size; (void)ws_size;
    const float* emb    = (const float*)d_in[0];
    const int*   target = (const int*)d_in[1];
    const int    P      = in_sizes[1] / NBATCH;   // 400*800 = 320000

    float* ws      = (float*)d_ws;
    float* sums    = ws + OFF_SUMS;
    float* counts  = ws + OFF_COUNTS;
    float* means   = ws + OFF_MEANS;
    float* hinge   = ws + OFF_HINGE;
    float* partial = ws + OFF_PART;
    float* out     = (float*)d_out;

    zero_ws_kernel<<<(WS_FLOATS + 255) / 256, 256, 0, stream>>>(ws, WS_FLOATS);
    seg_sum_kernel<<<dim3(256, NBATCH), 256, 0, stream>>>(emb, target, sums, counts, P);
    means_dist_kernel<<<NBATCH, 384, 0, stream>>>(sums, counts, means, partial);
    var_kernel<<<dim3(512, NBATCH), 256, 0, stream>>>(emb, target, means, hinge, P);
    finalize_kernel<<<1, 128, 0, stream>>>(hinge, counts, partial, out);
}